// SimpleGNN_16381005267207
// MI455X (gfx1250) — compile-verified
//
#include <hip/hip_runtime.h>
#include <hip/hip_bf16.h>

typedef __attribute__((ext_vector_type(2))) float v2f;
typedef __attribute__((ext_vector_type(8))) float v8f;

#define HID 64
#define NGRAPH 64
#define OUTD 10
#define WPAD 66   // even pad: keeps 8B alignment for ds_load_b64 B-fragments

// ---------------------------------------------------------------- zero scratch
__global__ void k_zero(float* __restrict__ p, size_t n) {
    size_t i = (size_t)blockIdx.x * blockDim.x + threadIdx.x;
    size_t s = (size_t)gridDim.x * blockDim.x;
    for (; i < n; i += s) p[i] = 0.0f;
}

// ------------------------------------------------- degree count (edge -> dst)
__global__ void k_deg(const int* __restrict__ edst, float* __restrict__ deg, int n_edges) {
    int e = blockIdx.x * blockDim.x + threadIdx.x;
    if (e < n_edges) atomicAdd(&deg[edst[e]], 1.0f);
}

// ----------------------------------------------------- per-graph node counts
__global__ void k_cnt(const int* __restrict__ batch, float* __restrict__ gcnt, int n_nodes) {
    int n = blockIdx.x * blockDim.x + threadIdx.x;
    if (n < n_nodes) atomicAdd(&gcnt[batch[n]], 1.0f);
}

// --------------------------------------------------- dinv = rsqrt(deg + 1)
__global__ void k_dinv(float* __restrict__ dinv, int n_nodes) {
    int n = blockIdx.x * blockDim.x + threadIdx.x;
    if (n < n_nodes) dinv[n] = rsqrtf(dinv[n] + 1.0f);
}

// --------------------------------------------------- h1 = x @ W1   (K = 3)
__global__ __launch_bounds__(256) void k_xw1(const float* __restrict__ x,
                                             const float* __restrict__ W1,
                                             float* __restrict__ h1, int n_nodes) {
    __shared__ float sW[3 * HID];
    if (threadIdx.x < 3 * HID) sW[threadIdx.x] = W1[threadIdx.x];
    __syncthreads();
    int idx = blockIdx.x * 256 + threadIdx.x;
    int n = idx >> 6, j = idx & 63;
    if (n < n_nodes) {
        float v = x[n * 3 + 0] * sW[j]
                + x[n * 3 + 1] * sW[HID + j]
                + x[n * 3 + 2] * sW[2 * HID + j];
        h1[idx] = v;
    }
}

// --------------- edge scatter: agg[dst] += norm * h[src]; one wave per edge
__global__ __launch_bounds__(256) void k_scatter(const float* __restrict__ h,
                                                 const int* __restrict__ esrc,
                                                 const int* __restrict__ edst,
                                                 const float* __restrict__ dinv,
                                                 float* __restrict__ agg, int n_edges) {
    int lane = threadIdx.x & 31;
    int eidx = blockIdx.x * 8 + (threadIdx.x >> 5);
    if (eidx >= n_edges) return;
    int s = esrc[eidx];
    int d = edst[eidx];
    float norm = dinv[s] * dinv[d];
    const float2* hp = (const float2*)(h + (size_t)s * HID);
    float2 v = hp[lane];                       // 256B coalesced per edge
    float* ap = agg + (size_t)d * HID + lane * 2;
    atomicAdd(ap + 0, v.x * norm);
    atomicAdd(ap + 1, v.y * norm);
}

// ------- out = relu(agg + h*dinv^2 + b)  (may alias agg for in-place update)
__global__ __launch_bounds__(256) void k_fin_relu(const float* __restrict__ agg,
                                                  const float* __restrict__ h,
                                                  const float* __restrict__ dinv,
                                                  const float* __restrict__ b,
                                                  float* __restrict__ out, int n_nodes) {
    int idx = blockIdx.x * 256 + threadIdx.x;
    int n = idx >> 6, j = idx & 63;
    if (n < n_nodes) {
        float di = dinv[n];
        float v = agg[idx] + h[idx] * di * di + b[j];
        out[idx] = fmaxf(v, 0.0f);
    }
}

// -------- layer-2 finalize fused with mean-pool accumulation into gsum[64,64]
__global__ __launch_bounds__(256) void k_fin_pool(const float* __restrict__ agg,
                                                  const float* __restrict__ h,
                                                  const float* __restrict__ dinv,
                                                  const float* __restrict__ b,
                                                  const int* __restrict__ batch,
                                                  float* __restrict__ gsum, int n_nodes) {
    int idx = blockIdx.x * 256 + threadIdx.x;
    int n = idx >> 6, j = idx & 63;
    if (n < n_nodes) {
        float di = dinv[n];
        float v = fmaxf(agg[idx] + h[idx] * di * di + b[j], 0.0f);
        atomicAdd(&gsum[batch[n] * HID + j], v);
    }
}

// ---------------------- h2 = out1 @ W2  via V_WMMA_F32_16X16X4_F32 (exact f32)
// 8 waves per block; each wave owns a 16-row panel and all four 16-col tiles.
// A rows are CLAMPED (not predicated) so loads stay unconditional and EXEC
// stays all-1s; out-of-range rows only feed output rows that are never stored.
__global__ __launch_bounds__(256) void k_gemm_wmma(const float* __restrict__ A,   // [N,64]
                                                   const float* __restrict__ W,   // [64,64]
                                                   float* __restrict__ C,         // [N,64]
                                                   int n_nodes) {
    // W transposed into LDS: sWt[n*WPAD + k] = W[k][n]; K-adjacent pairs are
    // contiguous + 8B aligned -> each B fragment is one ds_load_b64.
    __shared__ float sWt[HID * WPAD];
    for (int i = threadIdx.x; i < HID * HID; i += 256) {
        int k = i >> 6, n = i & 63;
        sWt[n * WPAD + k] = W[i];
    }
    __syncthreads();

    const int wave = threadIdx.x >> 5;
    const int lane = threadIdx.x & 31;
    const int row0 = (blockIdx.x * 8 + wave) * 16;
    if (row0 >= n_nodes) return;                 // wave-uniform: EXEC stays all-1s

    const int m     = lane & 15;                 // A-frag row / B-frag col
    const int khalf = (lane >> 4) * 2;           // lanes 16-31 carry K+2,K+3
    int row = row0 + m;
    if (row >= n_nodes) row = n_nodes - 1;       // clamp (store loop guards rows)

    const float* Arow = A + (size_t)row * HID + khalf;
    const float* B0   = &sWt[( 0 + m) * WPAD + khalf];
    const float* B1   = &sWt[(16 + m) * WPAD + khalf];
    const float* B2   = &sWt[(32 + m) * WPAD + khalf];
    const float* B3   = &sWt[(48 + m) * WPAD + khalf];

    v8f acc0 = {}, acc1 = {}, acc2 = {}, acc3 = {};

    #pragma unroll
    for (int k0 = 0; k0 < HID; k0 += 4) {
        v2f a  = *(const v2f*)(Arow + k0);       // one global_load_b64
        v2f b0 = *(const v2f*)(B0 + k0);         // one ds_load_b64 each
        v2f b1 = *(const v2f*)(B1 + k0);
        v2f b2 = *(const v2f*)(B2 + k0);
        v2f b3 = *(const v2f*)(B3 + k0);
        acc0 = __builtin_amdgcn_wmma_f32_16x16x4_f32(false, a, false, b0, (short)0, acc0, false, false);
        acc1 = __builtin_amdgcn_wmma_f32_16x16x4_f32(false, a, false, b1, (short)0, acc1, false, false);
        acc2 = __builtin_amdgcn_wmma_f32_16x16x4_f32(false, a, false, b2, (short)0, acc2, false, false);
        acc3 = __builtin_amdgcn_wmma_f32_16x16x4_f32(false, a, false, b3, (short)0, acc3, false, false);
    }

    // C/D layout: lane holds col (lane&15); VGPR r holds row r + 8*(lane>>4)
    const int half = lane >> 4;
    const int coln = lane & 15;
    #pragma unroll
    for (int r = 0; r < 8; ++r) {
        int mr = row0 + r + 8 * half;
        if (mr < n_nodes) {
            size_t base = (size_t)mr * HID;
            C[base +  0 + coln] = acc0[r];
            C[base + 16 + coln] = acc1[r];
            C[base + 32 + coln] = acc2[r];
            C[base + 48 + coln] = acc3[r];
        }
    }
}

// ----------------------------------- out[g] = (gsum[g]/cnt[g]) @ Wfc + bfc
__global__ __launch_bounds__(64) void k_fc(const float* __restrict__ gsum,
                                           const float* __restrict__ gcnt,
                                           const float* __restrict__ Wfc,
                                           const float* __restrict__ bfc,
                                           float* __restrict__ out) {
    __shared__ float mean[HID];
    int g = blockIdx.x;
    float c = fmaxf(gcnt[g], 1.0f);
    mean[threadIdx.x] = gsum[g * HID + threadIdx.x] / c;
    __syncthreads();
    if (threadIdx.x < OUTD) {
        float acc = bfc[threadIdx.x];
        #pragma unroll
        for (int h = 0; h < HID; ++h) acc += mean[h] * Wfc[h * OUTD + threadIdx.x];
        out[g * OUTD + threadIdx.x] = acc;
    }
}

extern "C" void kernel_launch(void* const* d_in, const int* in_sizes, int n_in,
                              void* d_out, int out_size, void* d_ws, size_t ws_size,
                              hipStream_t stream) {
    const float* x    = (const float*)d_in[0];
    const int*   ei   = (const int*)d_in[1];    // [2, E]
    const int*   batch= (const int*)d_in[2];
    const float* W1   = (const float*)d_in[3];
    const float* b1   = (const float*)d_in[4];
    const float* W2   = (const float*)d_in[5];
    const float* b2   = (const float*)d_in[6];
    const float* Wfc  = (const float*)d_in[7];
    const float* bfc  = (const float*)d_in[8];
    float*       out  = (float*)d_out;

    const int N = in_sizes[0] / 3;              // nodes
    const int E = in_sizes[1] / 2;              // edges
    const int* esrc = ei;
    const int* edst = ei + E;
    const size_t N64 = (size_t)N * HID;

    // workspace layout
    float* h1   = (float*)d_ws;
    float* h2   = h1 + N64;
    float* a1   = h2 + N64;                     // zeroed region starts here
    float* a2   = a1 + N64;
    float* dinv = a2 + N64;                     // doubles as deg accumulator
    float* gsum = dinv + N;
    float* gcnt = gsum + NGRAPH * HID;
    const size_t zero_n = 2 * N64 + (size_t)N + NGRAPH * HID + NGRAPH;

    const int elem_blocks = (int)((N64 + 255) / 256);
    const int edge_blocks = (E + 7) / 8;        // 8 edges (waves) per block

    // 0) zero scratch accumulators (a1, a2, deg/dinv, gsum, gcnt)
    k_zero<<<1024, 256, 0, stream>>>(a1, zero_n);
    // 1) degree counts + graph counts + dinv
    k_deg <<<(E + 255) / 256, 256, 0, stream>>>(edst, dinv, E);
    k_cnt <<<(N + 255) / 256, 256, 0, stream>>>(batch, gcnt, N);
    k_dinv<<<(N + 255) / 256, 256, 0, stream>>>(dinv, N);
    // 2) layer 1: transform, aggregate, finalize (relu in-place into a1)
    k_xw1     <<<elem_blocks, 256, 0, stream>>>(x, W1, h1, N);
    k_scatter <<<edge_blocks, 256, 0, stream>>>(h1, esrc, edst, dinv, a1, E);
    k_fin_relu<<<elem_blocks, 256, 0, stream>>>(a1, h1, dinv, b1, a1, N);
    // 3) layer 2: WMMA transform, aggregate, finalize fused with pooling
    k_gemm_wmma<<<(N / 16 + 7) / 8, 256, 0, stream>>>(a1, W2, h2, N);
    k_scatter  <<<edge_blocks, 256, 0, stream>>>(h2, esrc, edst, dinv, a2, E);
    k_fin_pool <<<elem_blocks, 256, 0, stream>>>(a2, h2, dinv, b2, batch, gsum, N);
    // 4) mean + FC head
    k_fc<<<NGRAPH, 64, 0, stream>>>(gsum, gcnt, Wfc, bfc, out);
}